// E3SchNet_53171695125406
// MI455X (gfx1250) — compile-verified
//
#include <hip/hip_runtime.h>

#define N_ATOMS 16000
#define N_EDGES 256000
#define FDIM    128
#define NFEAT   32
#define NLAYER  3
#define NRBF    20
#define EPW     16   // edges per wave in the edge kernel

__constant__ const float INV_SQRT_F   = 0.088388347648318447f;  // 1/sqrt(128)
__constant__ const float INV_SQRT_2NF = 0.125f;                 // 1/sqrt(64)
#define INV_SQ3f  0.57735026918962576f
#define SQ3f      1.7320508075688772f
#define LOG2f     0.69314718055994531f
#define PIf       3.14159265358979323f

typedef float v2f __attribute__((ext_vector_type(2)));
typedef float v8f __attribute__((ext_vector_type(8)));
typedef unsigned int u32x4 __attribute__((ext_vector_type(4)));
typedef int i32x4 __attribute__((ext_vector_type(4)));
typedef int i32x8 __attribute__((ext_vector_type(8)));

#if __has_builtin(__builtin_amdgcn_tensor_load_to_lds) && \
    __has_builtin(__builtin_amdgcn_s_wait_tensorcnt)
#define USE_TDM 1
#else
#define USE_TDM 0
#endif

// D = A(16x4 f32) * B(4x16 f32) + C, fp32 WMMA (exact vs fp32 reference).
// A frag: lane m = lane&15 holds K = 2*(lane>>4) + {0,1} in {a.x,a.y}
// B frag: lane n = lane&15 holds K = 2*(lane>>4) + {0,1} in {b.x,b.y}
// C/D:    VGPR r -> row (r + 8*(lane>>4)), col = lane&15
__device__ __forceinline__ v8f wmma4(v2f a, v2f b, v8f c) {
  return __builtin_amdgcn_wmma_f32_16x16x4_f32(
      /*neg_a=*/false, a, /*neg_b=*/false, b,
      /*c_mod=*/(short)0, c, /*reuse_a=*/false, /*reuse_b=*/false);
}

__device__ __forceinline__ float ssp(float x) {
  // softplus(x) - log(2), numerically guarded
  float sp = (x > 20.0f) ? x : log1pf(__expf(x));
  return sp - LOG2f;
}

// ---------------------------------------------------------------- geometry
__global__ __launch_bounds__(256) void k_geom(const float* __restrict__ r,
                                              float* __restrict__ fij,
                                              float* __restrict__ rcut,
                                              float* __restrict__ Yv) {
  int e = blockIdx.x * 256 + threadIdx.x;
  if (e >= N_EDGES) return;
  float x = r[3 * e + 0], y = r[3 * e + 1], z = r[3 * e + 2];
  float d = sqrtf(x * x + y * y + z * z);
  const float delta = 5.0f / 19.0f;          // linspace(0,5,20) spacing
  const float coeff = -0.5f / (delta * delta);
#pragma unroll
  for (int k = 0; k < NRBF; ++k) {
    float dd = d - delta * (float)k;
    fij[(size_t)e * NRBF + k] = __expf(coeff * dd * dd);
  }
  rcut[e] = (d < 5.0f) ? 0.5f * (__cosf(PIf * d * 0.2f) + 1.0f) : 0.0f;
  float inv = SQ3f / fmaxf(d, 1e-9f);
  Yv[3 * e + 0] = x * inv;
  Yv[3 * e + 1] = y * inv;
  Yv[3 * e + 2] = z * inv;
}

// ------------------------------------------------- xs = emb[Z] @ W_pe * c
__global__ __launch_bounds__(256) void k_embed(const int* __restrict__ Z,
                                               const float* __restrict__ emb,
                                               const float* __restrict__ Wpe,
                                               float* __restrict__ xs) {
  int wave = threadIdx.x >> 5, lane = threadIdx.x & 31;
  int mrow = lane & 15, khalf = lane >> 4;
  int m0 = blockIdx.x * 16, n0 = wave * 16;
  const float* Arow = emb + (size_t)Z[m0 + mrow] * FDIM;
  v8f acc = {};
  for (int k = 0; k < FDIM; k += 4) {
    int kk = k + 2 * khalf;
    v2f a = {Arow[kk], Arow[kk + 1]};
    v2f b = {Wpe[(size_t)kk * FDIM + n0 + mrow],
             Wpe[(size_t)(kk + 1) * FDIM + n0 + mrow]};
    acc = wmma4(a, b, acc);
  }
#pragma unroll
  for (int r = 0; r < 8; ++r)
    xs[(size_t)(m0 + r + 8 * khalf) * FDIM + n0 + mrow] = acc[r] * INV_SQRT_F;
}

// ------------------------- H = [xs@W0s | xv_x@W0v | xv_y@W0v | xv_z@W0v] * c
__global__ __launch_bounds__(256) void k_h(const float* __restrict__ xs,
                                           const float* __restrict__ xv,
                                           const float* __restrict__ W0s,
                                           const float* __restrict__ W0v,
                                           float* __restrict__ H) {
  int wave = threadIdx.x >> 5, lane = threadIdx.x & 31;
  int mrow = lane & 15, khalf = lane >> 4;
  int ch = wave >> 1, nt = wave & 1;
  int m0 = blockIdx.x * 16;
  const float* A = (ch == 0) ? xs : (xv + (size_t)(ch - 1) * N_ATOMS * FDIM);
  const float* B = (ch == 0) ? W0s : W0v;  // [128][32]
  const float* Arow = A + (size_t)(m0 + mrow) * FDIM;
  v8f acc = {};
  for (int k = 0; k < FDIM; k += 4) {
    int kk = k + 2 * khalf;
    v2f a = {Arow[kk], Arow[kk + 1]};
    v2f b = {B[(size_t)kk * NFEAT + nt * 16 + mrow],
             B[(size_t)(kk + 1) * NFEAT + nt * 16 + mrow]};
    acc = wmma4(a, b, acc);
  }
#pragma unroll
  for (int r = 0; r < 8; ++r)
    H[(size_t)(m0 + r + 8 * khalf) * FDIM + ch * 32 + nt * 16 + mrow] =
        acc[r] * INV_SQRT_F;
}

// ----------------------------------------------------------- edge kernel
__device__ __forceinline__ void edge_flush(float* __restrict__ agg, int i,
                                           int lane, float aS0, float aS1,
                                           const float aV0[3],
                                           const float aV1[3]) {
  float* base = agg + (size_t)i * 256;
  atomicAdd(base + lane, aS0);
  atomicAdd(base + 32 + lane, aS1);
#pragma unroll
  for (int d = 0; d < 3; ++d) {
    atomicAdd(base + 64 + lane * 3 + d, aV0[d]);          // t1a block (f=c)
    atomicAdd(base + 64 + (32 + lane) * 3 + d, aV1[d]);   // t1b block (f=32+c)
  }
}

// one wave per run of EPW consecutive edges; lane = channel (0..31).
// Filter-MLP weights staged in LDS (TDM for the 16KB Wf2 live sub-tile).
__global__ __launch_bounds__(256) void k_edge(
    const int* __restrict__ idx_i, const int* __restrict__ idx_j,
    const float* __restrict__ H, const float* __restrict__ fij,
    const float* __restrict__ rcut, const float* __restrict__ Yv,
    const float* __restrict__ Wf1, const float* __restrict__ bf1,
    const float* __restrict__ Wf2, const float* __restrict__ bf2,
    float* __restrict__ agg) {
  __shared__ float W2L[NFEAT * 128];    // live cols 0..127 of Wf2 (16 KB)
  __shared__ float Wf1L[NRBF * NFEAT];  // 2.5 KB
  __shared__ float bf1L[NFEAT];
  __shared__ float bf2L[128];

  // ---- stage small tables cooperatively
  for (int t = threadIdx.x; t < NRBF * NFEAT; t += 256) Wf1L[t] = Wf1[t];
  if (threadIdx.x < NFEAT) bf1L[threadIdx.x] = bf1[threadIdx.x];
  if (threadIdx.x < 128) bf2L[threadIdx.x] = bf2[threadIdx.x];

#if USE_TDM
  // ---- stage Wf2[32 rows][cols 0..127] via Tensor Data Mover (wave 0 only).
  // D# per CDNA5 ISA: 2D tile 128x32 from a tensor with
  // row stride 192 elements, data_size=4B, packed contiguously into LDS.
  if ((threadIdx.x >> 5) == 0) {
    uint64_t gaddr = (uint64_t)(uintptr_t)Wf2;
    uint32_t laddr = (uint32_t)(uintptr_t)W2L;  // generic LDS addr low 32b
    u32x4 g0;
    g0[0] = 1u;                                   // count=1, user mode
    g0[1] = laddr;                                // lds_addr
    g0[2] = (uint32_t)gaddr;                      // global_addr[31:0]
    g0[3] = (uint32_t)(gaddr >> 32) & 0x01FFFFFFu;
    g0[3] |= (2u << 30);                          // type=2 ("image")
    i32x8 g1;
    g1[0] = (int)(2u << 16);                      // data_size=2 (4 bytes)
    g1[1] = (int)((192u & 0xFFFFu) << 16);        // tensor_dim0[15:0]
    g1[2] = (int)((192u >> 16) | ((32u & 0xFFFFu) << 16));   // dim0 hi | dim1 lo
    g1[3] = (int)((32u >> 16) | (128u << 16));    // dim1 hi | tile_dim0=128
    g1[4] = (int)(32u);                           // tile_dim1=32, tile_dim2=0
    g1[5] = (int)(192u);                          // tensor_dim0_stride lo
    g1[6] = (int)((6144u & 0xFFFFu) << 16);       // d0s hi=0 | dim1_stride lo
    g1[7] = (int)(6144u >> 16);                   // dim1_stride hi
    i32x4 z4 = {0, 0, 0, 0};                      // groups 2/3 unused (2D)
    i32x8 z8 = {0, 0, 0, 0, 0, 0, 0, 0};          // extra group (clang-23 ABI)
    __builtin_amdgcn_tensor_load_to_lds(g0, g1, z4, z4, z8, 0);
    __builtin_amdgcn_s_wait_tensorcnt(0);
  }
#else
  for (int t = threadIdx.x; t < NFEAT * 128; t += 256)
    W2L[t] = Wf2[(size_t)(t >> 7) * 192 + (t & 127)];
#endif
  __syncthreads();

  int wid = blockIdx.x * (blockDim.x >> 5) + (threadIdx.x >> 5);
  int lane = threadIdx.x & 31;
  int e0 = wid * EPW;
  float aS0 = 0.f, aS1 = 0.f;
  float aV0[3] = {0.f, 0.f, 0.f}, aV1[3] = {0.f, 0.f, 0.f};
  int cur = idx_i[e0];
  for (int e = e0; e < e0 + EPW; ++e) {
    int i = idx_i[e];
    if (i != cur) {
      edge_flush(agg, cur, lane, aS0, aS1, aV0, aV1);
      aS0 = aS1 = 0.f;
      aV0[0] = aV0[1] = aV0[2] = 0.f;
      aV1[0] = aV1[1] = aV1[2] = 0.f;
      cur = i;
    }
    int j = idx_j[e];
    const float* Hj = H + (size_t)j * FDIM;
    float sj = Hj[lane];
    float vx = Hj[32 + lane], vy = Hj[64 + lane], vz = Hj[96 + lane];
    // prefetch next edge's gathered row (global_prefetch_b8)
    if (e + 1 < N_EDGES)
      __builtin_prefetch(H + (size_t)idx_j[e + 1] * FDIM + lane, 0, 3);
    float Yx = Yv[3 * e], Yy = Yv[3 * e + 1], Yz = Yv[3 * e + 2];
    float rc = rcut[e];
    // filter MLP: g = ssp(f_ij @ Wf1 + bf1)   (cross-lane via shfl, LDS wgts)
    float f = (lane < NRBF) ? fij[(size_t)e * NRBF + lane] : 0.f;
    float g = bf1L[lane];
#pragma unroll
    for (int r = 0; r < NRBF; ++r)
      g = fmaf(__shfl(f, r, 32), Wf1L[r * NFEAT + lane], g);
    g = ssp(g);
    // W = (g @ Wf2 + bf2) * rcut ; only the 4 live blocks (t1e/t2e are dead)
    float w0a = 0.f, w0b = 0.f, w1a = 0.f, w1b = 0.f;
#pragma unroll
    for (int r = 0; r < NFEAT; ++r) {
      float gr = __shfl(g, r, 32);
      const float* Wr = W2L + r * 128;
      w0a = fmaf(gr, Wr[lane], w0a);
      w0b = fmaf(gr, Wr[32 + lane], w0b);
      w1a = fmaf(gr, Wr[64 + lane], w1a);
      w1b = fmaf(gr, Wr[96 + lane], w1b);
    }
    w0a = (w0a + bf2L[lane]) * rc;
    w0b = (w0b + bf2L[32 + lane]) * rc;
    w1a = (w1a + bf2L[64 + lane]) * rc;
    w1b = (w1b + bf2L[96 + lane]) * rc;
    float t0b = (vx * Yx + vy * Yy + vz * Yz) * INV_SQ3f;
    aS0 += sj * w0a;
    aS1 += t0b * w0b;
    float sw = sj * w1a;
    aV0[0] += sw * Yx;  aV0[1] += sw * Yy;  aV0[2] += sw * Yz;
    aV1[0] += vx * w1b; aV1[1] += vy * w1b; aV1[2] += vz * w1b;
  }
  edge_flush(agg, cur, lane, aS0, aS1, aV0, aV1);
}

// -------------------------- atom update: fused (A -> ssp -> B) via LDS
__global__ __launch_bounds__(256) void k_update(
    const float* __restrict__ agg, const float* __restrict__ As,
    const float* __restrict__ Av, const float* __restrict__ Bs,
    const float* __restrict__ Bv, float* __restrict__ xs,
    float* __restrict__ xv) {
  __shared__ float inter[16 * FDIM];
  int wave = threadIdx.x >> 5, lane = threadIdx.x & 31;
  int mrow = lane & 15, khalf = lane >> 4;
  int m0 = blockIdx.x * 16, n0 = wave * 16;
  for (int ch = 0; ch < 4; ++ch) {
    const float* A1 = (ch == 0) ? As : Av;  // [64][128]
    const float* B2 = (ch == 0) ? Bs : Bv;  // [128][128]
    const float* arow =
        agg + (size_t)(m0 + mrow) * 256 + ((ch == 0) ? 0 : 64 + (ch - 1));
    int astride = (ch == 0) ? 1 : 3;
    // stage 1: inter = act( in(16x64) @ A1 * INV_SQRT_2NF )
    v8f acc = {};
    for (int k = 0; k < 64; k += 4) {
      int kk = k + 2 * khalf;
      v2f a = {arow[kk * astride], arow[(kk + 1) * astride]};
      v2f b = {A1[(size_t)kk * FDIM + n0 + mrow],
               A1[(size_t)(kk + 1) * FDIM + n0 + mrow]};
      acc = wmma4(a, b, acc);
    }
#pragma unroll
    for (int r = 0; r < 8; ++r) {
      float v = acc[r] * INV_SQRT_2NF;
      if (ch == 0) v = ssp(v);
      inter[(r + 8 * khalf) * FDIM + n0 + mrow] = v;
    }
    __syncthreads();
    // stage 2: out += inter(16x128) @ B2 * INV_SQRT_F
    v8f acc2 = {};
    for (int k = 0; k < FDIM; k += 4) {
      int kk = k + 2 * khalf;
      v2f a = {inter[mrow * FDIM + kk], inter[mrow * FDIM + kk + 1]};
      v2f b = {B2[(size_t)kk * FDIM + n0 + mrow],
               B2[(size_t)(kk + 1) * FDIM + n0 + mrow]};
      acc2 = wmma4(a, b, acc2);
    }
    float* out = (ch == 0) ? xs : (xv + (size_t)(ch - 1) * N_ATOMS * FDIM);
#pragma unroll
    for (int r = 0; r < 8; ++r)
      out[(size_t)(m0 + r + 8 * khalf) * FDIM + n0 + mrow] +=
          acc2[r] * INV_SQRT_F;
    __syncthreads();
  }
}

// -------------------------------------------------------------- launcher
extern "C" void kernel_launch(void* const* d_in, const int* in_sizes, int n_in,
                              void* d_out, int out_size, void* d_ws,
                              size_t ws_size, hipStream_t stream) {
  const int*   Z     = (const int*)d_in[0];
  const float* r_ij  = (const float*)d_in[1];
  const int*   idx_i = (const int*)d_in[2];
  const int*   idx_j = (const int*)d_in[3];
  const float* emb   = (const float*)d_in[4];
  const float* W_pe  = (const float*)d_in[5];
  const float* W0s   = (const float*)d_in[6];
  const float* W0v   = (const float*)d_in[7];
  const float* Wf1   = (const float*)d_in[8];
  const float* bf1   = (const float*)d_in[9];
  const float* Wf2   = (const float*)d_in[10];
  const float* bf2   = (const float*)d_in[11];
  const float* As    = (const float*)d_in[12];
  const float* Av    = (const float*)d_in[13];
  const float* Bs    = (const float*)d_in[14];
  const float* Bv    = (const float*)d_in[15];
  float* xs = (float*)d_out;

  // workspace carve (floats): xv(3 planes), H, agg, f_ij, rcut, Yv  (~74 MB)
  float* ws = (float*)d_ws;
  size_t o = 0;
  float* xv  = ws + o; o += (size_t)3 * N_ATOMS * FDIM;
  float* H   = ws + o; o += (size_t)N_ATOMS * FDIM;
  float* agg = ws + o; o += (size_t)N_ATOMS * 256;
  float* fij = ws + o; o += (size_t)N_EDGES * NRBF;
  float* rct = ws + o; o += (size_t)N_EDGES;
  float* Yv  = ws + o; o += (size_t)N_EDGES * 3;
  (void)o; (void)ws_size; (void)in_sizes; (void)n_in; (void)out_size;

  (void)hipMemsetAsync(xv, 0, (size_t)3 * N_ATOMS * FDIM * sizeof(float),
                       stream);
  k_geom<<<(N_EDGES + 255) / 256, 256, 0, stream>>>(r_ij, fij, rct, Yv);
  k_embed<<<N_ATOMS / 16, 256, 0, stream>>>(Z, emb, W_pe, xs);

  for (int l = 0; l < NLAYER; ++l) {
    k_h<<<N_ATOMS / 16, 256, 0, stream>>>(
        xs, xv, W0s + (size_t)l * FDIM * NFEAT, W0v + (size_t)l * FDIM * NFEAT,
        H);
    (void)hipMemsetAsync(agg, 0, (size_t)N_ATOMS * 256 * sizeof(float),
                         stream);
    k_edge<<<N_EDGES / (EPW * 8), 256, 0, stream>>>(
        idx_i, idx_j, H, fij, rct, Yv, Wf1 + (size_t)l * NRBF * NFEAT,
        bf1 + (size_t)l * NFEAT, Wf2 + (size_t)l * NFEAT * 192,
        bf2 + (size_t)l * 192, agg);
    k_update<<<N_ATOMS / 16, 256, 0, stream>>>(
        agg, As + (size_t)l * 64 * FDIM, Av + (size_t)l * 64 * FDIM,
        Bs + (size_t)l * FDIM * FDIM, Bv + (size_t)l * FDIM * FDIM, xs, xv);
  }
}